// TransformerAttentionModule_21062519619908
// MI455X (gfx1250) — compile-verified
//
#include <hip/hip_runtime.h>
#include <math.h>

// ---------------------------------------------------------------------------
// Graph transformer attention (GAT-style) for MI455X / gfx1250.
//   DIM=128, NUM_HEADS=8, HEAD_DIM=16, N=50000, E=800000.
// Pipeline:
//   1) q/k/v = x @ W{q,k,v}^T + b     -> fp32 WMMA 16x16x4 (native f32 matrix op)
//   2) scores[e,h] = <q[src],k[tgt]>/4 ; atomic segment-max over tgt
//   3) p = exp(score - m[tgt]) ; atomic segment-sum over tgt
//   4) agg[tgt] += v[src] * p/s[tgt]  (f32 global atomics, L2-resident)
//   5) out = agg @ Wo^T + bo          -> fp32 WMMA 16x16x4
// ---------------------------------------------------------------------------

typedef float v2f __attribute__((ext_vector_type(2)));
typedef float v8f __attribute__((ext_vector_type(8)));

#define DIM 128
#define NHEADS 8
#define HDIM 16

// ---------------------------------------------------------------------------
// GEMM: Y[M,128] = X[M,128] @ W^T[128,128] + bias.  One wave per 16x16 tile.
// A fragment: lane l holds x[tile_m*16 + (l&15)][k0 + 2*(l>>4) + {0,1}]
// B fragment: lane l holds W[tile_n*16 + (l&15)][k0 + 2*(l>>4) + {0,1}]
// D: vgpr r -> row (r + 8*(l>>4)), col (l&15).
// ---------------------------------------------------------------------------
__global__ void gemm128_wmma_f32(const float* __restrict__ X,
                                 const float* __restrict__ W,
                                 const float* __restrict__ bias,
                                 float* __restrict__ Y,
                                 int M)
{
    const int gt     = blockIdx.x * blockDim.x + threadIdx.x;
    const int wave   = gt >> 5;
    const int lane   = gt & 31;
    const int tile_m = wave >> 3;          // 8 tiles along N (128/16)
    const int tile_n = wave & 7;
    if (tile_m * 16 + 16 > M) return;

    const int half = lane >> 4;            // 0: K={0,1}, 1: K={2,3}
    const int r    = lane & 15;            // row (A) / col (B) within tile

    const float* __restrict__ xrow = X + (size_t)(tile_m * 16 + r) * DIM;
    const float* __restrict__ wrow = W + (size_t)(tile_n * 16 + r) * DIM;

    v8f acc = {};
#pragma unroll
    for (int k0 = 0; k0 < DIM; k0 += 4) {
        const int kk = k0 + 2 * half;
        v2f a = *(const v2f*)(xrow + kk);   // contiguous float2, 8B aligned
        v2f b = *(const v2f*)(wrow + kk);
        acc = __builtin_amdgcn_wmma_f32_16x16x4_f32(
                  /*neg_a=*/false, a, /*neg_b=*/false, b,
                  /*c_mod=*/(short)0, acc, /*reuse_a=*/false, /*reuse_b=*/false);
    }

    const int col = tile_n * 16 + r;
    const float bc = bias[col];
#pragma unroll
    for (int vr = 0; vr < 8; ++vr) {
        const int row = tile_m * 16 + vr + 8 * half;
        Y[(size_t)row * DIM + col] = acc[vr] + bc;
    }
}

// ---------------------------------------------------------------------------
// Init: m = -inf, s = 0, agg = 0.
// ---------------------------------------------------------------------------
__global__ void init_buffers(float* __restrict__ m, float* __restrict__ s,
                             float* __restrict__ agg, int nm, int nagg)
{
    const int i = blockIdx.x * blockDim.x + threadIdx.x;
    if (i < nm)  { m[i] = -INFINITY; s[i] = 0.0f; }
    if (i < nagg) agg[i] = 0.0f;
}

// Order-preserving float atomic max via integer punning (init is -inf).
__device__ __forceinline__ void atomicMaxF32(float* addr, float val)
{
    if (val >= 0.0f)
        atomicMax((int*)addr, __float_as_int(val));
    else
        atomicMin((unsigned int*)addr, (unsigned int)__float_as_int(val));
}

// ---------------------------------------------------------------------------
// Pass 1: per (edge, head) score + segment max.  One thread per (e,h).
// ---------------------------------------------------------------------------
__global__ void edge_scores_max(const float* __restrict__ q,
                                const float* __restrict__ k,
                                const int*   __restrict__ tgt,
                                const int*   __restrict__ src,
                                float* __restrict__ scores,
                                float* __restrict__ m,
                                int E)
{
    const int i = blockIdx.x * blockDim.x + threadIdx.x;
    if (i >= E * NHEADS) return;
    const int e = i >> 3;
    const int h = i & 7;
    const int sn = src[e];
    const int tn = tgt[e];

    const float4* qp = (const float4*)(q + (size_t)sn * DIM + h * HDIM);
    const float4* kp = (const float4*)(k + (size_t)tn * DIM + h * HDIM);
    float acc = 0.0f;
#pragma unroll
    for (int j = 0; j < 4; ++j) {
        const float4 a = qp[j];
        const float4 b = kp[j];
        acc += a.x * b.x + a.y * b.y + a.z * b.z + a.w * b.w;
    }
    acc *= 0.25f;                          // 1/sqrt(HEAD_DIM)
    scores[i] = acc;
    atomicMaxF32(m + (size_t)tn * NHEADS + h, acc);
}

// ---------------------------------------------------------------------------
// Pass 2: p = exp(score - m[tgt]); segment sum.  scores[] rewritten with p.
// ---------------------------------------------------------------------------
__global__ void edge_exp_sum(const int* __restrict__ tgt,
                             const float* __restrict__ m,
                             float* __restrict__ scores,
                             float* __restrict__ s,
                             int E)
{
    const int i = blockIdx.x * blockDim.x + threadIdx.x;
    if (i >= E * NHEADS) return;
    const int e = i >> 3;
    const int h = i & 7;
    const int tn = tgt[e];
    const float p = expf(scores[i] - m[(size_t)tn * NHEADS + h]);
    scores[i] = p;
    atomicAdd(s + (size_t)tn * NHEADS + h, p);
}

// ---------------------------------------------------------------------------
// Pass 3: agg[tgt] += v[src] * p / s[tgt].  One wave (32 lanes) per edge,
// each lane handles 4 contiguous floats (float4 gather + 4 f32 atomics).
// ---------------------------------------------------------------------------
__global__ void edge_aggregate(const float* __restrict__ v,
                               const int*   __restrict__ tgt,
                               const int*   __restrict__ src,
                               const float* __restrict__ p,
                               const float* __restrict__ s,
                               float* __restrict__ agg,
                               int E)
{
    const int t = blockIdx.x * blockDim.x + threadIdx.x;
    const int e = t >> 5;
    if (e >= E) return;
    const int lane = t & 31;
    const int off  = lane * 4;             // 0..124, head = off/16
    const int h    = off >> 4;
    const int sn = src[e];
    const int tn = tgt[e];

    const float w = p[(size_t)e * NHEADS + h] / s[(size_t)tn * NHEADS + h];
    const float4 vv = *(const float4*)(v + (size_t)sn * DIM + off);
    float* ap = agg + (size_t)tn * DIM + off;
    atomicAdd(ap + 0, vv.x * w);
    atomicAdd(ap + 1, vv.y * w);
    atomicAdd(ap + 2, vv.z * w);
    atomicAdd(ap + 3, vv.w * w);
}

// ---------------------------------------------------------------------------
// Launch
// inputs: 0=x 1=edge_index(2,E) 2=Wq 3=bq 4=Wk 5=bk 6=Wv 7=bv 8=Wo 9=bo
// ---------------------------------------------------------------------------
extern "C" void kernel_launch(void* const* d_in, const int* in_sizes, int n_in,
                              void* d_out, int out_size, void* d_ws, size_t ws_size,
                              hipStream_t stream)
{
    const float* x  = (const float*)d_in[0];
    const int*   ei = (const int*)d_in[1];
    const float* Wq = (const float*)d_in[2];
    const float* bq = (const float*)d_in[3];
    const float* Wk = (const float*)d_in[4];
    const float* bk = (const float*)d_in[5];
    const float* Wv = (const float*)d_in[6];
    const float* bv = (const float*)d_in[7];
    const float* Wo = (const float*)d_in[8];
    const float* bo = (const float*)d_in[9];
    float* out = (float*)d_out;

    const int N = in_sizes[0] / DIM;
    const int E = in_sizes[1] / 2;
    const int* tgt = ei;          // edge_index[0]
    const int* src = ei + E;      // edge_index[1]

    // Workspace layout (floats)
    float* q      = (float*)d_ws;
    float* kbuf   = q     + (size_t)N * DIM;
    float* vbuf   = kbuf  + (size_t)N * DIM;
    float* scores = vbuf  + (size_t)N * DIM;      // E*8 (reused as p)
    float* mbuf   = scores + (size_t)E * NHEADS;  // N*8
    float* sbuf   = mbuf  + (size_t)N * NHEADS;   // N*8
    float* agg    = sbuf  + (size_t)N * NHEADS;   // N*128

    const int BLK = 256;

    // init m/s/agg
    {
        const int n = N * DIM;                    // covers agg (> N*8)
        init_buffers<<<(n + BLK - 1) / BLK, BLK, 0, stream>>>(
            mbuf, sbuf, agg, N * NHEADS, n);
    }

    // q/k/v projections (fp32 WMMA)
    {
        const int waves   = (N / 16) * 8;
        const int threads = waves * 32;
        const int blocks  = (threads + BLK - 1) / BLK;
        gemm128_wmma_f32<<<blocks, BLK, 0, stream>>>(x, Wq, bq, q,    N);
        gemm128_wmma_f32<<<blocks, BLK, 0, stream>>>(x, Wk, bk, kbuf, N);
        gemm128_wmma_f32<<<blocks, BLK, 0, stream>>>(x, Wv, bv, vbuf, N);
    }

    // edge passes
    {
        const int n = E * NHEADS;
        edge_scores_max<<<(n + BLK - 1) / BLK, BLK, 0, stream>>>(
            q, kbuf, tgt, src, scores, mbuf, E);
        edge_exp_sum<<<(n + BLK - 1) / BLK, BLK, 0, stream>>>(
            tgt, mbuf, scores, sbuf, E);
        const int na = E * 32;
        edge_aggregate<<<(na + BLK - 1) / BLK, BLK, 0, stream>>>(
            vbuf, tgt, src, scores, sbuf, agg, E);
    }

    // output projection (fp32 WMMA)
    {
        const int waves   = (N / 16) * 8;
        const int threads = waves * 32;
        const int blocks  = (threads + BLK - 1) / BLK;
        gemm128_wmma_f32<<<blocks, BLK, 0, stream>>>(agg, Wo, bo, out, N);
    }
}